// KnnMSE_69965017252449
// MI455X (gfx1250) — compile-verified
//
#include <hip/hip_runtime.h>

// Problem constants (match the reference)
#define B_   16
#define N1_  2048
#define N2_  2048
#define F_   64
#define STR_ 67          // 3 coords + 64 features per row
#define QT_  (N2_ / 16)  // query tiles per batch = 128
#define CT_  (N1_ / 16)  // candidate tiles = 128
#define WAVES_PER_BLOCK 8
#define NBLOCKS ((B_ * QT_) / WAVES_PER_BLOCK)  // 2048 waves / 8 = 256 blocks

typedef float v2f __attribute__((ext_vector_type(2)));
typedef float v8f __attribute__((ext_vector_type(8)));

__device__ __forceinline__ void top3_insert(float d, int idx,
                                            float& b0, float& b1, float& b2,
                                            int& i0, int& i1, int& i2) {
    // branchless 3-element sorted insert (compiles to v_cmp + v_cndmask chains)
    bool lt2 = d < b2, lt1 = d < b1, lt0 = d < b0;
    float nb2 = lt2 ? (lt1 ? b1 : d) : b2;  int ni2 = lt2 ? (lt1 ? i1 : idx) : i2;
    float nb1 = lt1 ? (lt0 ? b0 : d) : b1;  int ni1 = lt1 ? (lt0 ? i0 : idx) : i1;
    float nb0 = lt0 ? d : b0;               int ni0 = lt0 ? idx : i0;
    b0 = nb0; b1 = nb1; b2 = nb2; i0 = ni0; i1 = ni1; i2 = ni2;
}

__global__ __launch_bounds__(256) void knn_mse_main(
    const float* __restrict__ true_x,   // (B*N1, 67)
    const float* __restrict__ pred_x,   // (B*N2, 67)
    float* __restrict__ partial)        // (NBLOCKS,) block partial sums
{
    const int lane   = threadIdx.x & 31;
    const int wid    = threadIdx.x >> 5;
    const int gwave  = blockIdx.x * WAVES_PER_BLOCK + wid;  // 0..2047
    const int b      = gwave >> 7;       // / QT_
    const int qtile  = gwave & (QT_ - 1);
    const int qbase  = qtile * 16;
    const int qi     = lane & 15;
    const bool hi    = lane >= 16;

    const float* c1 = true_x + (size_t)b * N1_ * STR_;
    const float* c2 = pred_x + (size_t)b * N2_ * STR_;

    // ---- load this lane's query coords, build B-matrix (4x16) registers ----
    const float* qrow = c2 + (size_t)(qbase + qi) * STR_;
    const float qx = qrow[0], qy = qrow[1], qz = qrow[2];
    const float qnorm = qx * qx + qy * qy + qz * qz;

    // B col n = (-2qx, -2qy, -2qz, 1); K striped across lane halves:
    //   VGPR0: lanes0-15 -> K=0, lanes16-31 -> K=1
    //   VGPR1: lanes0-15 -> K=2, lanes16-31 -> K=3
    v2f bmat;
    bmat.x = hi ? (-2.0f * qy) : (-2.0f * qx);
    bmat.y = hi ? 1.0f         : (-2.0f * qz);

    float d0 = 3.4e38f, d1 = 3.4e38f, d2v = 3.4e38f;
    int   i0 = 0, i1 = 0, i2 = 0;

    // ---- sweep all candidate tiles: WMMA gives (||c||^2 - 2 q.c) per pair ----
    for (int tile = 0; tile < CT_; ++tile) {
        const int cand = tile * 16 + qi;  // A row (candidate) this lane loads
        const float* crow = c1 + (size_t)cand * STR_;
        const float cx = crow[0], cy = crow[1], cz = crow[2];
        const float cn = cx * cx + cy * cy + cz * cz;

        // A (16x4): lanes0-15 hold K=0/1, lanes16-31 hold K=2/3 (ISA 7.12.2)
        v2f amat;
        amat.x = hi ? cz : cx;   // K=2 : K=0
        amat.y = hi ? cn : cy;   // K=3 : K=1

        v8f cacc = {};
        v8f dmat = __builtin_amdgcn_wmma_f32_16x16x4_f32(
            false, amat, false, bmat, (short)0, cacc, false, false);

#pragma unroll
        for (int r = 0; r < 8; ++r) {
            // D row M = r (+8 for hi half), col N = lane%16 (this lane's query)
            float dd = qnorm + dmat[r];
            int   ix = tile * 16 + r + (hi ? 8 : 0);
            top3_insert(dd, ix, d0, d1, d2v, i0, i1, i2);
        }
    }

    // ---- merge with partner lane (covers the other 8 rows of every tile) ----
    {
        float pb0 = __shfl_xor(d0, 16, 32), pb1 = __shfl_xor(d1, 16, 32),
              pb2 = __shfl_xor(d2v, 16, 32);
        int   pi0 = __shfl_xor(i0, 16, 32), pi1 = __shfl_xor(i1, 16, 32),
              pi2 = __shfl_xor(i2, 16, 32);
        top3_insert(pb0, pi0, d0, d1, d2v, i0, i1, i2);
        top3_insert(pb1, pi1, d0, d1, d2v, i0, i1, i2);
        top3_insert(pb2, pi2, d0, d1, d2v, i0, i1, i2);
    }

    // ---- inverse-distance-weighted interpolation + squared error ----
    // both lanes of a pair hold identical top-3; split the 64 features 32/32
    const float w0 = 1.0f / fmaxf(d0, 1e-16f);
    const float w1 = 1.0f / fmaxf(d1, 1e-16f);
    const float w2 = 1.0f / fmaxf(d2v, 1e-16f);
    const float winv = 1.0f / (w0 + w1 + w2);

    const int fs = hi ? 32 : 0;
    const float* n0 = c1 + (size_t)i0 * STR_ + 3 + fs;
    const float* n1 = c1 + (size_t)i1 * STR_ + 3 + fs;
    const float* n2 = c1 + (size_t)i2 * STR_ + 3 + fs;
    const float* fq = c2 + (size_t)(qbase + qi) * STR_ + 3 + fs;

    float acc = 0.0f;
#pragma unroll 4
    for (int f = 0; f < 32; ++f) {
        float interp = (w0 * n0[f] + w1 * n1[f] + w2 * n2[f]) * winv;
        float e = interp - fq[f];
        acc = fmaf(e, e, acc);
    }

    // ---- deterministic block reduction: wave shfl, then LDS across 8 waves ----
    for (int off = 16; off > 0; off >>= 1)
        acc += __shfl_down(acc, off, 32);

    __shared__ float sred[WAVES_PER_BLOCK];
    if (lane == 0) sred[wid] = acc;
    __syncthreads();
    if (threadIdx.x == 0) {
        float s = 0.0f;
#pragma unroll
        for (int w = 0; w < WAVES_PER_BLOCK; ++w) s += sred[w];
        partial[blockIdx.x] = s;
    }
}

__global__ __launch_bounds__(256) void knn_mse_reduce(
    const float* __restrict__ partial, float* __restrict__ out)
{
    __shared__ float s[NBLOCKS];
    const int t = threadIdx.x;
    s[t] = partial[t];
    __syncthreads();
    for (int off = NBLOCKS / 2; off > 0; off >>= 1) {
        if (t < off) s[t] += s[t + off];
        __syncthreads();
    }
    if (t == 0)
        out[0] = s[0] * (1.0f / ((float)B_ * (float)N2_ * (float)F_));
}

extern "C" void kernel_launch(void* const* d_in, const int* in_sizes, int n_in,
                              void* d_out, int out_size, void* d_ws, size_t ws_size,
                              hipStream_t stream) {
    const float* true_x = (const float*)d_in[0];
    const float* pred_x = (const float*)d_in[1];
    // d_in[2]/d_in[3] (batch1/batch2) are structured: row b*N -> batch b; unused.
    float* out     = (float*)d_out;
    float* partial = (float*)d_ws;   // NBLOCKS floats of scratch

    knn_mse_main<<<NBLOCKS, 256, 0, stream>>>(true_x, pred_x, partial);
    knn_mse_reduce<<<1, NBLOCKS, 0, stream>>>(partial, out);
}